// Net_59339268161711
// MI455X (gfx1250) — compile-verified
//
#include <hip/hip_runtime.h>
#include <hip/hip_bf16.h>
#include <math.h>

typedef __attribute__((ext_vector_type(2))) float v2f;
typedef __attribute__((ext_vector_type(8))) float v8f;

// ---------------------------------------------------------------------------
// Degree / normalization
// ---------------------------------------------------------------------------
__global__ void init_deg_kernel(float* __restrict__ deg, int n) {
  int i = blockIdx.x * blockDim.x + threadIdx.x;
  if (i < n) deg[i] = 1.0f;  // self loop
}

__global__ void count_deg_kernel(const int* __restrict__ col, float* __restrict__ deg, int e) {
  int i = blockIdx.x * blockDim.x + threadIdx.x;
  if (i < e) atomicAdd(&deg[col[i]], 1.0f);
}

__global__ void dis_kernel(float* __restrict__ deg, int n) {
  int i = blockIdx.x * blockDim.x + threadIdx.x;
  if (i < n) deg[i] = 1.0f / sqrtf(deg[i]);  // deg >= 1 always
}

// ---------------------------------------------------------------------------
// GEMM1: xw[N,16] = x[N,128] @ W1[128,16] via V_WMMA_F32_16X16X4_F32
// One wave computes one 16x16 output tile (all 16 output cols), K=128 in
// 32 steps of 4. A layout (ISA 7.12.2, 32-bit A 16x4): lanes 0-15 hold
// K=k0,k0+1; lanes 16-31 hold K=k0+2,k0+3 for row M=lane%16. B mirrored.
// ---------------------------------------------------------------------------
__global__ __launch_bounds__(128) void gemm1_wmma_kernel(const float* __restrict__ x,
                                                         const float* __restrict__ W1,
                                                         float* __restrict__ xw,
                                                         int n_tiles) {
  int wave = threadIdx.x >> 5;
  int tile = blockIdx.x * 4 + wave;
  if (tile >= n_tiles) return;  // wave-granular guard: EXEC all-ones inside
  int lane = threadIdx.x & 31;
  int m = lane & 15;
  int half = lane >> 4;

  const float* xr = x + ((size_t)tile * 16 + m) * 128;
  v8f c = {};
#pragma unroll
  for (int k0 = 0; k0 < 128; k0 += 4) {
    int ka = k0 + half * 2;
    v2f a;
    a.x = xr[ka];
    a.y = xr[ka + 1];
    v2f b;
    b.x = W1[ka * 16 + m];
    b.y = W1[(ka + 1) * 16 + m];
    c = __builtin_amdgcn_wmma_f32_16x16x4_f32(false, a, false, b, (short)0, c, false, false);
  }
  // D layout: VGPR r -> row M = r + 8*half, col N = m
  float* yr = xw + ((size_t)tile * 16 + half * 8) * 16 + m;
#pragma unroll
  for (int r = 0; r < 8; ++r) yr[(size_t)r * 16] = c[r];
}

// ---------------------------------------------------------------------------
// Layer-1 propagate: acc1 = self-loop init, then edge scatter-add.
// ---------------------------------------------------------------------------
__global__ void init_acc1_kernel(const float* __restrict__ xw, const float* __restrict__ dis,
                                 float* __restrict__ acc1, int n) {
  int i = blockIdx.x * blockDim.x + threadIdx.x;  // over N*16
  if (i < n) {
    int node = i >> 4;
    float d = dis[node];
    acc1[i] = xw[i] * d * d;
  }
}

__global__ void scatter1_kernel(const int* __restrict__ row, const int* __restrict__ col,
                                const float* __restrict__ xw, const float* __restrict__ dis,
                                float* __restrict__ acc1, long long total) {
  long long gid = (long long)blockIdx.x * blockDim.x + threadIdx.x;  // over E*16
  if (gid >= total) return;
  int e = (int)(gid >> 4);
  int f = (int)(gid & 15);
  int r = row[e];
  int cix = col[e];
  float nrm = dis[r] * dis[cix];
  atomicAdd(&acc1[(size_t)cix * 16 + f], xw[(size_t)r * 16 + f] * nrm);
}

// ---------------------------------------------------------------------------
// GEMM2: hw2[N,8] = relu(acc1 + b1)[N,16] @ W2[16,8]  (W2 zero-padded to 16
// cols inside the B operand). Bias+ReLU fused into the A-operand load.
// ---------------------------------------------------------------------------
__global__ __launch_bounds__(128) void gemm2_wmma_kernel(const float* __restrict__ acc1,
                                                         const float* __restrict__ b1,
                                                         const float* __restrict__ W2,
                                                         float* __restrict__ hw2,
                                                         int n_tiles) {
  int wave = threadIdx.x >> 5;
  int tile = blockIdx.x * 4 + wave;
  if (tile >= n_tiles) return;
  int lane = threadIdx.x & 31;
  int m = lane & 15;
  int half = lane >> 4;

  const float* ar = acc1 + ((size_t)tile * 16 + m) * 16;
  v8f c = {};
#pragma unroll
  for (int k0 = 0; k0 < 16; k0 += 4) {
    int ka = k0 + half * 2;
    v2f a;
    a.x = fmaxf(ar[ka] + b1[ka], 0.0f);
    a.y = fmaxf(ar[ka + 1] + b1[ka + 1], 0.0f);
    v2f b;
    b.x = (m < 8) ? W2[ka * 8 + m] : 0.0f;
    b.y = (m < 8) ? W2[(ka + 1) * 8 + m] : 0.0f;
    c = __builtin_amdgcn_wmma_f32_16x16x4_f32(false, a, false, b, (short)0, c, false, false);
  }
  if (m < 8) {  // predicated stores AFTER all WMMAs: legal
    float* yr = hw2 + ((size_t)tile * 16 + half * 8) * 8 + m;
#pragma unroll
    for (int r = 0; r < 8; ++r) yr[(size_t)r * 8] = c[r];
  }
}

// ---------------------------------------------------------------------------
// Layer-2 propagate + sigmoid finalize.
// ---------------------------------------------------------------------------
__global__ void init_acc2_kernel(const float* __restrict__ hw2, const float* __restrict__ dis,
                                 float* __restrict__ acc2, int n) {
  int i = blockIdx.x * blockDim.x + threadIdx.x;  // over N*8
  if (i < n) {
    int node = i >> 3;
    float d = dis[node];
    acc2[i] = hw2[i] * d * d;
  }
}

__global__ void scatter2_kernel(const int* __restrict__ row, const int* __restrict__ col,
                                const float* __restrict__ hw2, const float* __restrict__ dis,
                                float* __restrict__ acc2, long long total) {
  long long gid = (long long)blockIdx.x * blockDim.x + threadIdx.x;  // over E*8
  if (gid >= total) return;
  int e = (int)(gid >> 3);
  int f = (int)(gid & 7);
  int r = row[e];
  int cix = col[e];
  float nrm = dis[r] * dis[cix];
  atomicAdd(&acc2[(size_t)cix * 8 + f], hw2[(size_t)r * 8 + f] * nrm);
}

__global__ void finalize_kernel(const float* __restrict__ acc2, const float* __restrict__ b2,
                                float* __restrict__ out, int n) {
  int i = blockIdx.x * blockDim.x + threadIdx.x;  // over N*8
  if (i < n) {
    float v = acc2[i] + b2[i & 7];
    out[i] = 1.0f / (1.0f + expf(-v));
  }
}

// ---------------------------------------------------------------------------
// Host entry
// ---------------------------------------------------------------------------
extern "C" void kernel_launch(void* const* d_in, const int* in_sizes, int n_in,
                              void* d_out, int out_size, void* d_ws, size_t ws_size,
                              hipStream_t stream) {
  const float* x   = (const float*)d_in[0];   // [N,128]
  const int*   ei  = (const int*)d_in[1];     // [2,E] flat
  const float* W1  = (const float*)d_in[2];   // [128,16]
  const float* b1  = (const float*)d_in[3];   // [16]
  const float* W2  = (const float*)d_in[4];   // [16,8]
  const float* b2  = (const float*)d_in[5];   // [8]
  float* out = (float*)d_out;

  const int N = in_sizes[0] / 128;            // 100000
  const int E = in_sizes[1] / 2;              // 3200000
  const int* row = ei;                        // edge_index[0]
  const int* col = ei + E;                    // edge_index[1]

  // workspace layout (floats)
  float* ws   = (float*)d_ws;
  float* dis  = ws;                           // N   (deg, then 1/sqrt(deg) in place)
  float* xw   = dis + N;                      // N*16
  float* acc1 = xw + (size_t)N * 16;          // N*16
  float* hw2  = acc1 + (size_t)N * 16;        // N*8
  float* acc2 = hw2 + (size_t)N * 8;          // N*8

  const int B = 256;
  const int n_tiles = N / 16;                 // N % 16 == 0 for this problem

  // normalization
  init_deg_kernel<<<(N + B - 1) / B, B, 0, stream>>>(dis, N);
  count_deg_kernel<<<(E + B - 1) / B, B, 0, stream>>>(col, dis, E);
  dis_kernel<<<(N + B - 1) / B, B, 0, stream>>>(dis, N);

  // layer 1
  gemm1_wmma_kernel<<<(n_tiles + 3) / 4, 128, 0, stream>>>(x, W1, xw, n_tiles);
  init_acc1_kernel<<<((N * 16) + B - 1) / B, B, 0, stream>>>(xw, dis, acc1, N * 16);
  {
    long long total = (long long)E * 16;
    long long blocks = (total + B - 1) / B;
    scatter1_kernel<<<(unsigned)blocks, B, 0, stream>>>(row, col, xw, dis, acc1, total);
  }

  // layer 2
  gemm2_wmma_kernel<<<(n_tiles + 3) / 4, 128, 0, stream>>>(acc1, b1, W2, hw2, n_tiles);
  init_acc2_kernel<<<((N * 8) + B - 1) / B, B, 0, stream>>>(hw2, dis, acc2, N * 8);
  {
    long long total = (long long)E * 8;
    long long blocks = (total + B - 1) / B;
    scatter2_kernel<<<(unsigned)blocks, B, 0, stream>>>(row, col, hw2, dis, acc2, total);
  }
  finalize_kernel<<<((N * 8) + B - 1) / B, B, 0, stream>>>(acc2, b2, out, N * 8);
}